// PCPiecewiseRigid_19722489823913
// MI455X (gfx1250) — compile-verified
//
#include <hip/hip_runtime.h>
#include <hip/hip_bf16.h>

typedef float v2f __attribute__((ext_vector_type(2)));
typedef float v8f __attribute__((ext_vector_type(8)));

#define BB   32
#define NN   102400
#define TT   64
#define SS   1600
#define K3   (3 * NN)      // 307200
#define ROW6 (6 * NN)      // row stride of x (B,6,N)
#define HID  32
#define OUTD (6 * TT)      // 384
#define PI_F 3.14159265358979f

// ---------------------------------------------------------------------------
// Phase 1: hacc(32x32) += xflat(32 x 307200) @ w1(307200 x 32), split-K.
// 300 blocks x 8 waves; each wave covers 128 K in steps of 4 using
// V_WMMA_F32_16X16X4_F32 on a 2x2 grid of 16x16 output tiles.
// ---------------------------------------------------------------------------
__global__ __launch_bounds__(256) void gemm1_kernel(const float* __restrict__ x,
                                                    const float* __restrict__ w1,
                                                    float* __restrict__ hacc) {
    __shared__ float red[8][1024];
    const int lane = threadIdx.x & 31;
    const int wave = threadIdx.x >> 5;
    const int lo   = lane & 15;
    const int hi   = lane >> 4;
    const int kbase = blockIdx.x * 1024 + wave * 128;

    v8f c00 = {}, c01 = {}, c10 = {}, c11 = {};
    const float* xr0 = x + (size_t)lo * ROW6;        // A rows, M block 0
    const float* xr1 = x + (size_t)(16 + lo) * ROW6; // A rows, M block 1

#pragma unroll 4
    for (int s = 0; s < 32; ++s) {
        const int k  = kbase + 4 * s;
        const int ka = k + 2 * hi;                   // this lane's K pair
        // A fragments: 16x4 f32, one contiguous float2 per lane
        v2f a0 = *(const v2f*)(xr0 + ka);
        v2f a1 = *(const v2f*)(xr1 + ka);
        // B fragments: 4x16 f32, two strided scalars per lane
        v2f b0, b1v;
        b0.x  = w1[(size_t)ka * HID + lo];
        b0.y  = w1[(size_t)(ka + 1) * HID + lo];
        b1v.x = w1[(size_t)ka * HID + 16 + lo];
        b1v.y = w1[(size_t)(ka + 1) * HID + 16 + lo];
        __builtin_prefetch(xr0 + ka + 512, 0, 1);    // global_prefetch_b8
        c00 = __builtin_amdgcn_wmma_f32_16x16x4_f32(false, a0, false, b0,  (short)0, c00, false, false);
        c01 = __builtin_amdgcn_wmma_f32_16x16x4_f32(false, a0, false, b1v, (short)0, c01, false, false);
        c10 = __builtin_amdgcn_wmma_f32_16x16x4_f32(false, a1, false, b0,  (short)0, c10, false, false);
        c11 = __builtin_amdgcn_wmma_f32_16x16x4_f32(false, a1, false, b1v, (short)0, c11, false, false);
    }

    // Cross-wave reduction in LDS, then 1 atomic per C element per block.
    float* dst = &red[wave][lane * 32];
#pragma unroll
    for (int i = 0; i < 8; ++i) {
        dst[0 * 8 + i] = c00[i];
        dst[1 * 8 + i] = c01[i];
        dst[2 * 8 + i] = c10[i];
        dst[3 * 8 + i] = c11[i];
    }
    __syncthreads();
    for (int j = threadIdx.x; j < 1024; j += 256) {
        float sum = 0.f;
#pragma unroll
        for (int w = 0; w < 8; ++w) sum += red[w][j];
        const int l    = j >> 5;
        const int slot = j & 31;
        const int frag = slot >> 3;       // 0..3 = (mb, nb)
        const int i    = slot & 7;
        const int mb = frag >> 1, nb = frag & 1;
        const int lhi = l >> 4, llo = l & 15;
        const int row = mb * 16 + i + 8 * lhi;  // f32 C layout: VGPR i -> M=i / M=8+i
        const int col = nb * 16 + llo;
        atomicAdd(&hacc[row * 32 + col], sum);
    }
}

// ---------------------------------------------------------------------------
// Phase 2a: z(32x384) = tanh(relu(hacc + b1) @ w2 + b2)
// ---------------------------------------------------------------------------
__global__ void z_kernel(const float* __restrict__ hacc, const float* __restrict__ b1,
                         const float* __restrict__ w2, const float* __restrict__ b2,
                         float* __restrict__ z) {
    const int o = blockIdx.x * blockDim.x + threadIdx.x;
    if (o >= BB * OUTD) return;
    const int m = o / OUTD, col = o % OUTD;
    float acc = b2[col];
#pragma unroll
    for (int j = 0; j < HID; ++j)
        acc += fmaxf(hacc[m * HID + j] + b1[j], 0.f) * w2[j * OUTD + col];
    z[o] = tanhf(acc);
}

// ---------------------------------------------------------------------------
// Phase 2b: per (b,t): R = Z(a0)Y(a1)X(a2), alpha = PI*z[3:6], trans = z[3:6]
// T[k] = trans[k]*sz[k] + center[k] - sum_m center[m]*R[m][k]
// ---------------------------------------------------------------------------
__global__ void rigid_kernel(const float* __restrict__ z, const float* __restrict__ sz,
                             float* __restrict__ Rt) {
    const int idx = blockIdx.x * blockDim.x + threadIdx.x;
    if (idx >= BB * TT) return;
    const float* zz = z + (size_t)idx * 6;
    const float t0 = zz[3], t1 = zz[4], t2 = zz[5];
    const float a0 = PI_F * t0, a1 = PI_F * t1, a2 = PI_F * t2;
    const float c0 = cosf(a0), s0 = sinf(a0);
    const float c1 = cosf(a1), s1 = sinf(a1);
    const float c2 = cosf(a2), s2 = sinf(a2);
    float r[9];
    r[0] = c0 * c1; r[1] = -s0 * c2 + c0 * s1 * s2; r[2] =  s0 * s2 + c0 * s1 * c2;
    r[3] = s0 * c1; r[4] =  c0 * c2 + s0 * s1 * s2; r[5] = -c0 * s2 + s0 * s1 * c2;
    r[6] = -s1;     r[7] =  c1 * s2;                r[8] =  c1 * c2;
    const float sx = sz[0], sy = sz[1], szv = sz[2];
    const float cx = 0.5f * sx, cy = 0.5f * sy, cz = 0.5f * szv;
    float* o = Rt + (size_t)idx * 12;
#pragma unroll
    for (int i = 0; i < 9; ++i) o[i] = r[i];
    o[9]  = t0 * sx  + cx - (cx * r[0] + cy * r[3] + cz * r[6]);
    o[10] = t1 * sy  + cy - (cx * r[1] + cy * r[4] + cz * r[7]);
    o[11] = t2 * szv + cz - (cx * r[2] + cy * r[5] + cz * r[8]);
}

// ---------------------------------------------------------------------------
// Phase 3: out[b, j, n] = sum_k R[k][j] x[b,k,n] + T[j] for j<3 (R^T x + T),
//          out[b, 3:6, n] = x[b, 3:6, n].   tess is identity: t = n / S.
// float4-vectorized; segments (S=1600) are multiples of 4.
// ---------------------------------------------------------------------------
__global__ __launch_bounds__(256) void transform_kernel(const float* __restrict__ x,
                                                        const float* __restrict__ Rt,
                                                        float* __restrict__ out) {
    const int b     = blockIdx.x / (NN / 1024);
    const int chunk = blockIdx.x % (NN / 1024);
    const int n0 = chunk * 1024 + threadIdx.x * 4;
    const int t  = n0 / SS;
    const float* rt = Rt + ((size_t)b * TT + t) * 12;
    float r[12];
#pragma unroll
    for (int i = 0; i < 12; ++i) r[i] = rt[i];

    const float* xb = x + (size_t)b * ROW6;
    float*       ob = out + (size_t)b * ROW6;
    const float4 x0 = *(const float4*)(xb + 0 * NN + n0);
    const float4 x1 = *(const float4*)(xb + 1 * NN + n0);
    const float4 x2 = *(const float4*)(xb + 2 * NN + n0);

    float4 y0, y1, y2;
    y0.x = r[0]*x0.x + r[3]*x1.x + r[6]*x2.x + r[9];
    y0.y = r[0]*x0.y + r[3]*x1.y + r[6]*x2.y + r[9];
    y0.z = r[0]*x0.z + r[3]*x1.z + r[6]*x2.z + r[9];
    y0.w = r[0]*x0.w + r[3]*x1.w + r[6]*x2.w + r[9];
    y1.x = r[1]*x0.x + r[4]*x1.x + r[7]*x2.x + r[10];
    y1.y = r[1]*x0.y + r[4]*x1.y + r[7]*x2.y + r[10];
    y1.z = r[1]*x0.z + r[4]*x1.z + r[7]*x2.z + r[10];
    y1.w = r[1]*x0.w + r[4]*x1.w + r[7]*x2.w + r[10];
    y2.x = r[2]*x0.x + r[5]*x1.x + r[8]*x2.x + r[11];
    y2.y = r[2]*x0.y + r[5]*x1.y + r[8]*x2.y + r[11];
    y2.z = r[2]*x0.z + r[5]*x1.z + r[8]*x2.z + r[11];
    y2.w = r[2]*x0.w + r[5]*x1.w + r[8]*x2.w + r[11];

    *(float4*)(ob + 0 * NN + n0) = y0;
    *(float4*)(ob + 1 * NN + n0) = y1;
    *(float4*)(ob + 2 * NN + n0) = y2;
    // pass-through channels 3..5
    *(float4*)(ob + 3 * NN + n0) = *(const float4*)(xb + 3 * NN + n0);
    *(float4*)(ob + 4 * NN + n0) = *(const float4*)(xb + 4 * NN + n0);
    *(float4*)(ob + 5 * NN + n0) = *(const float4*)(xb + 5 * NN + n0);
}

// ---------------------------------------------------------------------------
// Phase 4a: pts_1[t] = mean over segment of A[:3]
// ---------------------------------------------------------------------------
__global__ void pts1_kernel(const float* __restrict__ A, float* __restrict__ p1) {
    __shared__ float sd[3 * 256];
    const int t = blockIdx.x, tid = threadIdx.x;
    float s0 = 0.f, s1 = 0.f, s2 = 0.f;
    for (int n = tid; n < SS; n += 256) {
        const int g = t * SS + n;
        s0 += A[0 * NN + g];
        s1 += A[1 * NN + g];
        s2 += A[2 * NN + g];
    }
    sd[tid] = s0; sd[256 + tid] = s1; sd[512 + tid] = s2;
    __syncthreads();
    for (int off = 128; off; off >>= 1) {
        if (tid < off) {
            sd[tid]       += sd[tid + off];
            sd[256 + tid] += sd[256 + tid + off];
            sd[512 + tid] += sd[512 + tid + off];
        }
        __syncthreads();
    }
    if (tid == 0) {
        const float inv = 1.f / SS;
        p1[t * 3 + 0] = sd[0] * inv;
        p1[t * 3 + 1] = sd[256] * inv;
        p1[t * 3 + 2] = sd[512] * inv;
    }
}

// ---------------------------------------------------------------------------
// Phase 4b: pts_2[b,t] = R^T * mean(x3 segment) + T  (mean commutes w/ affine)
// ---------------------------------------------------------------------------
__global__ void pts2_kernel(const float* __restrict__ x, const float* __restrict__ Rt,
                            float* __restrict__ p2) {
    __shared__ float sd[3 * 256];
    const int b = blockIdx.x >> 6, t = blockIdx.x & 63, tid = threadIdx.x;
    const float* xb = x + (size_t)b * ROW6 + t * SS;
    float s0 = 0.f, s1 = 0.f, s2 = 0.f;
    for (int n = tid; n < SS; n += 256) {
        s0 += xb[0 * NN + n];
        s1 += xb[1 * NN + n];
        s2 += xb[2 * NN + n];
    }
    sd[tid] = s0; sd[256 + tid] = s1; sd[512 + tid] = s2;
    __syncthreads();
    for (int off = 128; off; off >>= 1) {
        if (tid < off) {
            sd[tid]       += sd[tid + off];
            sd[256 + tid] += sd[256 + tid + off];
            sd[512 + tid] += sd[512 + tid + off];
        }
        __syncthreads();
    }
    if (tid == 0) {
        const float inv = 1.f / SS;
        const float m0 = sd[0] * inv, m1 = sd[256] * inv, m2 = sd[512] * inv;
        const float* r = Rt + (size_t)blockIdx.x * 12;
        p2[(size_t)blockIdx.x * 3 + 0] = r[0]*m0 + r[3]*m1 + r[6]*m2 + r[9];
        p2[(size_t)blockIdx.x * 3 + 1] = r[1]*m0 + r[4]*m1 + r[7]*m2 + r[10];
        p2[(size_t)blockIdx.x * 3 + 2] = r[2]*m0 + r[5]*m1 + r[8]*m2 + r[11];
    }
}

// ---------------------------------------------------------------------------
// Phase 4c: reg[b] = sum_{i,j} (mask[i,j]*(cdist(p1)[i,j]-cdist(p2_b)[i,j]))^2
// ---------------------------------------------------------------------------
__global__ void reg_kernel(const float* __restrict__ p1, const float* __restrict__ p2,
                           const float* __restrict__ mask, float* __restrict__ out) {
    __shared__ float sp1[TT * 3];
    __shared__ float sp2[TT * 3];
    __shared__ float sd[256];
    const int b = blockIdx.x, tid = threadIdx.x;
    if (tid < TT * 3) {
        sp1[tid] = p1[tid];
        sp2[tid] = p2[(size_t)b * TT * 3 + tid];
    }
    __syncthreads();
    float acc = 0.f;
    for (int idx = tid; idx < TT * TT; idx += 256) {
        const int i = idx >> 6, j = idx & 63;
        float dx = sp1[i*3+0] - sp1[j*3+0];
        float dy = sp1[i*3+1] - sp1[j*3+1];
        float dz = sp1[i*3+2] - sp1[j*3+2];
        const float d1 = sqrtf(dx*dx + dy*dy + dz*dz + 1e-12f);
        dx = sp2[i*3+0] - sp2[j*3+0];
        dy = sp2[i*3+1] - sp2[j*3+1];
        dz = sp2[i*3+2] - sp2[j*3+2];
        const float d2 = sqrtf(dx*dx + dy*dy + dz*dz + 1e-12f);
        const float v = mask[idx] * (d1 - d2);
        acc += v * v;
    }
    sd[tid] = acc;
    __syncthreads();
    for (int off = 128; off; off >>= 1) {
        if (tid < off) sd[tid] += sd[tid + off];
        __syncthreads();
    }
    if (tid == 0) out[(size_t)BB * ROW6 + b] = sd[0];
}

// ---------------------------------------------------------------------------
extern "C" void kernel_launch(void* const* d_in, const int* in_sizes, int n_in,
                              void* d_out, int out_size, void* d_ws, size_t ws_size,
                              hipStream_t stream) {
    const float* x    = (const float*)d_in[0];
    const float* w1   = (const float*)d_in[1];
    const float* b1   = (const float*)d_in[2];
    const float* w2   = (const float*)d_in[3];
    const float* b2   = (const float*)d_in[4];
    const float* A    = (const float*)d_in[5];
    const float* mask = (const float*)d_in[6];
    const float* sz   = (const float*)d_in[7];
    // d_in[8] = tess: identity permutation arange(N).reshape(T,S); folded in.
    float* out = (float*)d_out;

    float* ws   = (float*)d_ws;
    float* hacc = ws;            // 1024  floats: xflat @ w1 accumulator
    float* z    = ws + 1024;     // 12288 floats: tanh output (B x 384)
    float* Rt   = ws + 13312;    // 24576 floats: per (b,t) [R(9) | T(3)]
    float* p1   = ws + 37888;    // 192   floats: pts_1 (T x 3)
    float* p2   = ws + 38080;    // 6144  floats: pts_2 (B x T x 3)

    hipMemsetAsync(hacc, 0, 1024 * sizeof(float), stream);
    gemm1_kernel<<<K3 / 1024, 256, 0, stream>>>(x, w1, hacc);
    z_kernel<<<(BB * OUTD + 255) / 256, 256, 0, stream>>>(hacc, b1, w2, b2, z);
    rigid_kernel<<<(BB * TT + 255) / 256, 256, 0, stream>>>(z, sz, Rt);
    transform_kernel<<<BB * (NN / 1024), 256, 0, stream>>>(x, Rt, out);
    pts1_kernel<<<TT, 256, 0, stream>>>(A, p1);
    pts2_kernel<<<BB * TT, 256, 0, stream>>>(x, Rt, p2);
    reg_kernel<<<BB, 256, 0, stream>>>(p1, p2, mask, out);
}